// WaferLayout_16827681865914
// MI455X (gfx1250) — compile-verified
//
#include <hip/hip_runtime.h>

typedef float v2f __attribute__((ext_vector_type(2)));
typedef float v8f __attribute__((ext_vector_type(8)));

#define WAFER_R   150.0f
#define T_AMBIENT  25.0f
#define T_LIMIT    85.0f
#define ALPHA_C     0.01f
#define EPS_C       1e-6f
#define LAMBDA_COMM 1e-6f
#define INV_PI      0.3183098861837907f
// INV_2SIG2 * log2(e) = (1/(2*50^2)) * 1.4426950408889634
#define C_TH        2.8853900817779268e-4f

// ---------------------------------------------------------------- zero
__global__ void wl_zero(float* __restrict__ rowsum, float* __restrict__ acc, int S) {
    int i = blockIdx.x * blockDim.x + threadIdx.x;
    if (i < S) rowsum[i] = 0.0f;
    if (i < 4) acc[i] = 0.0f;
}

// ---------------------------------------------------------------- setup: r_chip, log2(tdp), L_boundary
__global__ void wl_setup(const float* __restrict__ pos, const float* __restrict__ area,
                         const float* __restrict__ tdp,
                         float* __restrict__ rc, float* __restrict__ lt,
                         float* __restrict__ acc, int S) {
    int i = blockIdx.x * blockDim.x + threadIdx.x;
    float v2 = 0.0f;
    if (i < S) {
        float x = pos[2 * i], y = pos[2 * i + 1];
        float n2 = x * x + y * y;
        float rchip = __builtin_amdgcn_sqrtf(area[i] * INV_PI + EPS_C);
        float rcen  = __builtin_amdgcn_sqrtf(n2 + EPS_C);
        float viol  = fmaxf(rcen + rchip - WAFER_R, 0.0f);
        v2 = viol * viol;
        rc[i] = rchip;
        lt[i] = __builtin_amdgcn_logf(tdp[i]);   // v_log_f32 = log2; folded into exp2 arg
    }
    for (int m = 1; m < 32; m <<= 1) v2 += __shfl_xor(v2, m, 32);
    if ((threadIdx.x & 31) == 0) atomicAdd(acc + 0, v2);
}

// ---------------------------------------------------------------- pairs
// One wave per 16x32 super-tile (two 16x16 j-tiles sharing the i-block).
// Rows (M) = j-block, cols (N) = i-block. Per j-tile, three V_WMMA_F32_16X16X4_F32:
//   r2m : r2_ij        = (-2xj)(xi) + (-2yj)(yi) + |pj|^2*1 + 1*|pi|^2
//   rsm : rc_j + rc_i  = rcj*1 + 1*rci
//   eam : -c'*r2 + log2(tdp_j)   (B operand shared with r2m)
// Epilogue is fully lane-local: ov = relu(rsm - sqrt(r2+eps)) masked i<j;
// kv = v_exp_f32(eam). Thermal row-sum over j == column sum here.
__device__ __forceinline__ void wl_tile_epilogue(const v8f& r2m, const v8f& rsm, const v8f& eam,
                                                 int gi, int jbase, float& ovacc, float& thacc) {
    #pragma unroll
    for (int v = 0; v < 8; ++v) {
        const int gj = jbase + v;
        const float r2 = fmaxf(r2m[v], 0.0f);
        const float d  = __builtin_amdgcn_sqrtf(r2 + EPS_C);
        const float ov = fmaxf(rsm[v] - d, 0.0f);
        if (gi < gj) ovacc = fmaf(ov, ov, ovacc);      // strict upper triangle of (i,j)
        thacc += __builtin_amdgcn_exp2f(eam[v]);       // = exp(-r2/(2s^2)) * tdp_j
    }
}

__global__ void wl_pairs(const float* __restrict__ pos,
                         const float* __restrict__ rc,
                         const float* __restrict__ lt,
                         float* __restrict__ rowsum,
                         float* __restrict__ acc, int nt2) {
    const int wave = threadIdx.x >> 5;
    const int lane = threadIdx.x & 31;
    const int wtile = blockIdx.x * (blockDim.x >> 5) + wave;
    const int ti = wtile / nt2, tp = wtile % nt2;
    const int i0 = ti << 4, j0 = tp << 5;      // two j-tiles: j0 and j0+16
    const int half = lane >> 4;                // 0: K=0,1 slots; 1: K=2,3 slots
    const int l = lane & 15;
    const bool h = (half != 0);

    const float2* p2 = (const float2*)pos;

    // ---- i-side (shared by both j-tiles, and by WMMA 1 & 3)
    const float2 pi = p2[i0 + l];
    const float n2i = pi.x * pi.x + pi.y * pi.y;
    const float rci = rc[i0 + l];
    v2f b1, b2;
    b1.x = h ? 1.0f : pi.x;   b1.y = h ? n2i : pi.y;
    b2.x = h ? 0.0f : 1.0f;   b2.y = h ? 0.0f : rci;

    // ---- j-side operands for the two tiles
    const float2 pja = p2[j0 + l];
    const float2 pjb = p2[j0 + 16 + l];
    const float n2ja = pja.x * pja.x + pja.y * pja.y;
    const float n2jb = pjb.x * pjb.x + pjb.y * pjb.y;
    const float rcja = rc[j0 + l],      rcjb = rc[j0 + 16 + l];
    const float ltja = lt[j0 + l],      ltjb = lt[j0 + 16 + l];

    v2f a1a, a2a, a3a, a1b, a2b, a3b;
    a1a.x = h ? n2ja : -2.0f * pja.x;  a1a.y = h ? 1.0f : -2.0f * pja.y;
    a1b.x = h ? n2jb : -2.0f * pjb.x;  a1b.y = h ? 1.0f : -2.0f * pjb.y;
    a2a.x = h ? 0.0f : rcja;           a2a.y = h ? 0.0f : 1.0f;
    a2b.x = h ? 0.0f : rcjb;           a2b.y = h ? 0.0f : 1.0f;
    a3a.x = h ? (ltja - C_TH * n2ja) : (2.0f * C_TH * pja.x);
    a3a.y = h ? (-C_TH)              : (2.0f * C_TH * pja.y);
    a3b.x = h ? (ltjb - C_TH * n2jb) : (2.0f * C_TH * pjb.x);
    a3b.y = h ? (-C_TH)              : (2.0f * C_TH * pjb.y);

    const v8f z = {};
    v8f r2a = __builtin_amdgcn_wmma_f32_16x16x4_f32(false, a1a, false, b1, (short)0, z, false, false);
    v8f r2b = __builtin_amdgcn_wmma_f32_16x16x4_f32(false, a1b, false, b1, (short)0, z, false, false);
    v8f rsa = __builtin_amdgcn_wmma_f32_16x16x4_f32(false, a2a, false, b2, (short)0, z, false, false);
    v8f rsb = __builtin_amdgcn_wmma_f32_16x16x4_f32(false, a2b, false, b2, (short)0, z, false, false);
    v8f eaa = __builtin_amdgcn_wmma_f32_16x16x4_f32(false, a3a, false, b1, (short)0, z, false, false);
    v8f eab = __builtin_amdgcn_wmma_f32_16x16x4_f32(false, a3b, false, b1, (short)0, z, false, false);

    const int gi = i0 + l;                         // this lane's column = global i
    const int jb = (half << 3);                    // row offset of this half
    float ovacc = 0.0f, thacc = 0.0f;
    wl_tile_epilogue(r2a, rsa, eaa, gi, j0 + jb,      ovacc, thacc);
    wl_tile_epilogue(r2b, rsb, eab, gi, j0 + 16 + jb, ovacc, thacc);

    // thermal: column sum for gi over 32 rows of j -> one atomic per lane 0..15
    thacc += __shfl_xor(thacc, 16, 32);
    if (lane < 16) atomicAdd(&rowsum[gi], thacc);

    // overlap: one atomic per wave (512 pair-elements)
    ovacc += __shfl_xor(ovacc, 1, 32);
    ovacc += __shfl_xor(ovacc, 2, 32);
    ovacc += __shfl_xor(ovacc, 4, 32);
    ovacc += __shfl_xor(ovacc, 8, 32);
    ovacc += __shfl_xor(ovacc, 16, 32);
    if (lane == 0) atomicAdd(acc + 1, ovacc);
}

// ---------------------------------------------------------------- comm
__global__ void wl_comm(const float* __restrict__ pos, const float* __restrict__ traffic,
                        const int* __restrict__ mapping, float* __restrict__ acc, int n) {
    int t = blockIdx.x * blockDim.x + threadIdx.x;
    float s = 0.0f;
    if (t < n) {
        int d1 = mapping[t], d2 = mapping[t + 1];
        if (d1 != d2) {
            float dx = pos[2 * d1]     - pos[2 * d2];
            float dy = pos[2 * d1 + 1] - pos[2 * d2 + 1];
            s = traffic[t] * __builtin_amdgcn_sqrtf(dx * dx + dy * dy + EPS_C);
        }
    }
    for (int m = 1; m < 32; m <<= 1) s += __shfl_xor(s, m, 32);
    if ((threadIdx.x & 31) == 0) atomicAdd(acc + 2, s);
}

// ---------------------------------------------------------------- finalize
__global__ void wl_final(const float* __restrict__ rowsum, const float* __restrict__ acc,
                         float* __restrict__ out, int S) {
    __shared__ float smax[32];
    float mx = -1e30f;
    for (int i = threadIdx.x; i < S; i += blockDim.x) mx = fmaxf(mx, rowsum[i]);
    for (int m = 16; m >= 1; m >>= 1) mx = fmaxf(mx, __shfl_xor(mx, m, 32));
    if ((threadIdx.x & 31) == 0) smax[threadIdx.x >> 5] = mx;
    __syncthreads();
    if (threadIdx.x < 32) {
        int nw = blockDim.x >> 5;
        float v = (threadIdx.x < (unsigned)nw) ? smax[threadIdx.x] : -1e30f;
        for (int m = 16; m >= 1; m >>= 1) v = fmaxf(v, __shfl_xor(v, m, 32));
        if (threadIdx.x == 0) {
            float tmax = T_AMBIENT + ALPHA_C * v;
            float lt = fmaxf(tmax - T_LIMIT, 0.0f);
            lt = lt * lt;
            float lb = acc[0], lo = acc[1], lc = acc[2];
            out[0] = lb + lo + LAMBDA_COMM * lc + lt;   // L_layout
            out[1] = lb;                                 // L_boundary
            out[2] = lo;                                 // L_overlap
            out[3] = lc;                                 // L_comm
            out[4] = lt;                                 // L_thermal
        }
    }
}

// ---------------------------------------------------------------- launch
extern "C" void kernel_launch(void* const* d_in, const int* in_sizes, int n_in,
                              void* d_out, int out_size, void* d_ws, size_t ws_size,
                              hipStream_t stream) {
    const float* pos     = (const float*)d_in[0];
    const float* area    = (const float*)d_in[1];
    const float* tdp     = (const float*)d_in[2];
    const float* traffic = (const float*)d_in[3];
    const int*   mapping = (const int*)d_in[4];
    const int S = in_sizes[1];        // 4096
    const int K = in_sizes[4];        // 8192

    float* ws     = (float*)d_ws;
    float* rc     = ws;               // S
    float* lt     = ws + S;           // S
    float* rowsum = ws + 2 * S;       // S
    float* acc    = ws + 3 * S;       // 4 accumulators
    float* out    = (float*)d_out;

    const int nt  = S / 16;           // 256 j/i tile count
    const int nt2 = nt / 2;           // 128 super-tiles per row
    const int waves = nt * nt2;       // 32768 waves, 8 per 256-thread block

    wl_zero <<<(S + 255) / 256, 256, 0, stream>>>(rowsum, acc, S);
    wl_setup<<<(S + 255) / 256, 256, 0, stream>>>(pos, area, tdp, rc, lt, acc, S);
    wl_pairs<<<waves / 8, 256, 0, stream>>>(pos, rc, lt, rowsum, acc, nt2);
    wl_comm <<<(K - 1 + 255) / 256, 256, 0, stream>>>(pos, traffic, mapping, acc, K - 1);
    wl_final<<<1, 1024, 0, stream>>>(rowsum, acc, out, S);
}